// GCN_Net_15702400434553
// MI455X (gfx1250) — compile-verified
//
#include <hip/hip_runtime.h>
#include <math.h>

// Problem constants (match reference)
#define N_NODES 100000
#define N_EDGES 1600000
#define F_IN    256
#define F_H     128
#define N_CLS   10
#define H_PAD   16       // padded class dim for WMMA tile

typedef float v2f __attribute__((ext_vector_type(2)));
typedef float v8f __attribute__((ext_vector_type(8)));

// ---------------------------------------------------------------- degree
__global__ __launch_bounds__(256) void k_deg_init(float* __restrict__ deg) {
    int i = blockIdx.x * 256 + threadIdx.x;
    if (i < N_NODES) deg[i] = 1.0f;                 // self-loop contributes 1
}

__global__ __launch_bounds__(256) void k_deg_add(const int* __restrict__ dst,
                                                 float* __restrict__ deg) {
    int e = blockIdx.x * 256 + threadIdx.x;
    if (e < N_EDGES) atomicAdd(&deg[dst[e]], 1.0f);
}

__global__ __launch_bounds__(256) void k_dinv(float* __restrict__ deg) {
    int i = blockIdx.x * 256 + threadIdx.x;
    if (i < N_NODES) {
        float d = deg[i];
        deg[i] = (d > 0.0f) ? rsqrtf(d) : 0.0f;     // deg >= 1 always, but match ref
    }
}

// ---------------------------------------------------------------- zero fill
__global__ __launch_bounds__(256) void k_zero(float* __restrict__ p, long n) {
    long i = (long)blockIdx.x * 256 + threadIdx.x;
    if (i < n) p[i] = 0.0f;
}

// ---------------------------------------------------------------- GEMM1: h1 = x @ W1  (N x 256) @ (256 x 128), fp32 WMMA
// One block (8 waves) per 16-row tile; wave w covers output columns [16w,16w+16).
#define LDS_STRIDE 260   // 256 + 4 pad -> conflict-free b64 LDS reads
__global__ __launch_bounds__(256) void k_gemm1(const float* __restrict__ x,
                                               const float* __restrict__ W1,
                                               float* __restrict__ h1) {
    __shared__ float As[16 * LDS_STRIDE];
    const int tile = blockIdx.x;               // 0 .. 6249  (100000/16)
    const int tid  = threadIdx.x;

    // Cooperative load of the 16x256 A tile (coalesced float4)
    const float4* xg = (const float4*)(x + (size_t)tile * 16 * F_IN);
    #pragma unroll
    for (int it = 0; it < 4; ++it) {
        int q   = tid + 256 * it;              // float4 index 0..1023
        int row = q >> 6;                      // 64 float4 per row
        int c4  = q & 63;
        float4 v = xg[row * 64 + c4];
        float* d = &As[row * LDS_STRIDE + c4 * 4];
        d[0] = v.x; d[1] = v.y; d[2] = v.z; d[3] = v.w;
    }
    __syncthreads();

    const int wave = tid >> 5;                 // 0..7 -> column tile
    const int lane = tid & 31;
    const int m    = lane & 15;
    const int kg   = lane >> 4;                // lane group selects K pair
    const int n0   = wave * 16;

    v8f c = {0.f,0.f,0.f,0.f,0.f,0.f,0.f,0.f};
    for (int k0 = 0; k0 < F_IN; k0 += 4) {
        int ka = k0 + 2 * kg;
        // A fragment: lane m holds A[m][ka], A[m][ka+1]
        v2f a = *(const v2f*)&As[m * LDS_STRIDE + ka];
        // B fragment: lane holds B[ka][n0+m], B[ka+1][n0+m]
        v2f b;
        b.x = W1[(size_t)ka       * F_H + n0 + m];
        b.y = W1[(size_t)(ka + 1) * F_H + n0 + m];
        c = __builtin_amdgcn_wmma_f32_16x16x4_f32(false, a, false, b,
                                                  (short)0, c, false, false);
    }

    // C layout: VGPR r -> row r (lanes 0-15) / row r+8 (lanes 16-31), col = n0+m
    float* out = h1 + (size_t)tile * 16 * F_H;
    #pragma unroll
    for (int r = 0; r < 8; ++r) {
        int row = r + 8 * kg;
        out[(size_t)row * F_H + n0 + m] = c[r];
    }
}

// ---------------------------------------------------------------- scatter layer 1
// 32 threads per edge, 4 features each (float4 gather, 4 float atomics)
__global__ __launch_bounds__(256) void k_scatter1(const int* __restrict__ src,
                                                  const int* __restrict__ dst,
                                                  const float* __restrict__ dinv,
                                                  const float* __restrict__ h1,
                                                  float* __restrict__ agg1) {
    long gid = (long)blockIdx.x * 256 + threadIdx.x;
    long e   = gid >> 5;
    if (e >= N_EDGES) return;
    int f4 = (int)(gid & 31);
    int s = src[e], d = dst[e];
    float nrm = dinv[s] * dinv[d];
    float4 v = *(const float4*)&h1[(size_t)s * F_H + f4 * 4];
    float* o = &agg1[(size_t)d * F_H + f4 * 4];
    atomicAdd(&o[0], v.x * nrm);
    atomicAdd(&o[1], v.y * nrm);
    atomicAdd(&o[2], v.z * nrm);
    atomicAdd(&o[3], v.w * nrm);
}

// self-loop + bias + ReLU (in place on agg1)
__global__ __launch_bounds__(256) void k_post1(const float* __restrict__ h1,
                                               const float* __restrict__ dinv,
                                               const float* __restrict__ b1,
                                               float* __restrict__ agg1) {
    long gid = (long)blockIdx.x * 256 + threadIdx.x;
    if (gid >= (long)N_NODES * F_H) return;
    int i = (int)(gid >> 7);
    int f = (int)(gid & (F_H - 1));
    float di = dinv[i];
    float v = agg1[gid] + h1[gid] * di * di + b1[f];
    agg1[gid] = v > 0.0f ? v : 0.0f;
}

// ---------------------------------------------------------------- W2 padding (128x10 -> 128x16, zero cols 10..15)
__global__ __launch_bounds__(256) void k_padW2(const float* __restrict__ W2,
                                               float* __restrict__ Bp) {
    int t = blockIdx.x * 256 + threadIdx.x;
    if (t >= F_H * H_PAD) return;
    int k = t >> 4, c = t & 15;
    Bp[t] = (c < N_CLS) ? W2[k * N_CLS + c] : 0.0f;
}

// ---------------------------------------------------------------- GEMM2: h2 = out1 @ W2pad  (N x 128) @ (128 x 16), fp32 WMMA
__global__ __launch_bounds__(256) void k_gemm2(const float* __restrict__ A,
                                               const float* __restrict__ Bp,
                                               float* __restrict__ h2) {
    const int wave = threadIdx.x >> 5;
    const int lane = threadIdx.x & 31;
    const int tile = blockIdx.x * 8 + wave;    // 16-row tile
    if ((size_t)tile * 16 >= N_NODES) return;  // wave-uniform: EXEC stays full
    const int m  = lane & 15;
    const int kg = lane >> 4;

    const float* Ar = A + ((size_t)tile * 16 + m) * F_H;
    v8f c = {0.f,0.f,0.f,0.f,0.f,0.f,0.f,0.f};
    for (int k0 = 0; k0 < F_H; k0 += 4) {
        int ka = k0 + 2 * kg;
        v2f a = *(const v2f*)&Ar[ka];
        v2f b;
        b.x = Bp[ka * H_PAD + m];
        b.y = Bp[(ka + 1) * H_PAD + m];
        c = __builtin_amdgcn_wmma_f32_16x16x4_f32(false, a, false, b,
                                                  (short)0, c, false, false);
    }
    float* out = h2 + (size_t)tile * 16 * H_PAD;
    #pragma unroll
    for (int r = 0; r < 8; ++r) {
        int row = r + 8 * kg;
        out[(size_t)row * H_PAD + m] = c[r];
    }
}

// ---------------------------------------------------------------- scatter layer 2 (10 valid features, padded to 16)
__global__ __launch_bounds__(256) void k_scatter2(const int* __restrict__ src,
                                                  const int* __restrict__ dst,
                                                  const float* __restrict__ dinv,
                                                  const float* __restrict__ h2,
                                                  float* __restrict__ agg2) {
    long gid = (long)blockIdx.x * 256 + threadIdx.x;
    long e   = gid >> 4;
    if (e >= N_EDGES) return;
    int f = (int)(gid & 15);
    if (f >= N_CLS) return;
    int s = src[e], d = dst[e];
    float nrm = dinv[s] * dinv[d];
    atomicAdd(&agg2[(size_t)d * H_PAD + f], h2[(size_t)s * H_PAD + f] * nrm);
}

// self-loop + bias + log_softmax
__global__ __launch_bounds__(256) void k_post2(const float* __restrict__ h2,
                                               const float* __restrict__ agg2,
                                               const float* __restrict__ dinv,
                                               const float* __restrict__ b2,
                                               float* __restrict__ out) {
    int i = blockIdx.x * 256 + threadIdx.x;
    if (i >= N_NODES) return;
    float di = dinv[i];
    float d2 = di * di;
    float v[N_CLS];
    float mx = -INFINITY;
    #pragma unroll
    for (int c = 0; c < N_CLS; ++c) {
        float t = agg2[(size_t)i * H_PAD + c] + h2[(size_t)i * H_PAD + c] * d2 + b2[c];
        v[c] = t;
        mx = fmaxf(mx, t);
    }
    float s = 0.0f;
    #pragma unroll
    for (int c = 0; c < N_CLS; ++c) s += expf(v[c] - mx);
    float lse = logf(s) + mx;
    #pragma unroll
    for (int c = 0; c < N_CLS; ++c) out[(size_t)i * N_CLS + c] = v[c] - lse;
}

// ---------------------------------------------------------------- launcher
extern "C" void kernel_launch(void* const* d_in, const int* in_sizes, int n_in,
                              void* d_out, int out_size, void* d_ws, size_t ws_size,
                              hipStream_t stream) {
    const float* x   = (const float*)d_in[0];                 // [N, 256]
    const int*   ei  = (const int*)d_in[1];                   // [2, E]
    const float* W1  = (const float*)d_in[2];                 // [256, 128]
    const float* b1  = (const float*)d_in[3];                 // [128]
    const float* W2  = (const float*)d_in[4];                 // [128, 10]
    const float* b2  = (const float*)d_in[5];                 // [10]
    float*       out = (float*)d_out;                         // [N, 10]

    const int* src = ei;
    const int* dst = ei + N_EDGES;

    // Workspace layout (floats). h2/agg2 alias the dead h1 buffer after post1.
    float* ws    = (float*)d_ws;
    float* dinv  = ws;                                        // N (also deg scratch)
    float* h1    = dinv + 100352;                             // N*128
    float* agg1  = h1 + (size_t)N_NODES * F_H;                // N*128
    float* W2pad = agg1 + (size_t)N_NODES * F_H;              // 128*16
    float* h2    = h1;                                        // N*16 (aliases h1)
    float* agg2  = h1 + (size_t)N_NODES * H_PAD;              // N*16 (aliases h1)

    const int nblk_node = (N_NODES + 255) / 256;

    // degrees + symmetric norm
    k_deg_init<<<nblk_node, 256, 0, stream>>>(dinv);
    k_deg_add <<<(N_EDGES + 255) / 256, 256, 0, stream>>>(dst, dinv);
    k_dinv    <<<nblk_node, 256, 0, stream>>>(dinv);

    // layer 1: dense WMMA GEMM, then normalized scatter, self-loop+bias+relu
    k_gemm1<<<N_NODES / 16, 256, 0, stream>>>(x, W1, h1);
    {
        long n = (long)N_NODES * F_H;
        k_zero<<<(unsigned)((n + 255) / 256), 256, 0, stream>>>(agg1, n);
    }
    {
        long t = (long)N_EDGES * 32;
        k_scatter1<<<(unsigned)((t + 255) / 256), 256, 0, stream>>>(src, dst, dinv, h1, agg1);
    }
    {
        long n = (long)N_NODES * F_H;
        k_post1<<<(unsigned)((n + 255) / 256), 256, 0, stream>>>(h1, dinv, b1, agg1);
    }

    // layer 2: pad W2, WMMA GEMM, scatter, self-loop+bias+log_softmax
    k_padW2<<<(F_H * H_PAD + 255) / 256, 256, 0, stream>>>(W2, W2pad);
    k_gemm2<<<(N_NODES / 16 + 7) / 8, 256, 0, stream>>>(agg1, W2pad, h2);
    {
        long n = (long)N_NODES * H_PAD;
        k_zero<<<(unsigned)((n + 255) / 256), 256, 0, stream>>>(agg2, n);
    }
    {
        long t = (long)N_EDGES * 16;
        k_scatter2<<<(unsigned)((t + 255) / 256), 256, 0, stream>>>(src, dst, dinv, h2, agg2);
    }
    k_post2<<<nblk_node, 256, 0, stream>>>(h2, agg2, dinv, b2, out);
}